// SingleTemporalExt_47991964566193
// MI455X (gfx1250) — compile-verified
//
#include <hip/hip_runtime.h>
#include <math.h>

// ---------------------------------------------------------------------------
// Problem constants (from reference)
// ---------------------------------------------------------------------------
constexpr int BSZ   = 512;
constexpr int SEQ   = 3;
constexpr int FF    = 5;
constexpr int CC    = 62;
constexpr int EE    = 5;
constexpr int DM    = CC * EE * FF;      // 1550
constexpr int ED    = 2 * DM;            // 3100
constexpr int DSTATE= 16;
constexpr int DTR   = (DM + 15) / 16;    // 97
constexpr int DBCW  = DTR + 2 * DSTATE;  // 129
constexpr int VOCAB = 128;
constexpr int OUTD  = 256;
constexpr int ROWS  = BSZ * SEQ;         // 1536
constexpr float EPS = 1e-5f;

typedef float v2f __attribute__((ext_vector_type(2)));
typedef float v8f __attribute__((ext_vector_type(8)));

#define DEV __device__ __forceinline__

DEV int imin(int a, int b) { return a < b ? a : b; }
DEV float softplusf(float x) { return x > 20.f ? x : log1pf(expf(x)); }
DEV float siluf(float x) { return x / (1.f + expf(-x)); }

DEV v8f wmma16x16x4(v2f a, v2f b, v8f c) {
  // D = A(16x4, f32) * B(4x16, f32) + C(16x16, f32)
  return __builtin_amdgcn_wmma_f32_16x16x4_f32(
      /*neg_a=*/false, a, /*neg_b=*/false, b,
      /*c_mod=*/(short)0, c, /*reuse_a=*/false, /*reuse_b=*/false);
}

// ---------------------------------------------------------------------------
// LDS-staged fp32 WMMA GEMM:  Cout[M,N] = act( A[M,K]·B[N,K]^T + bias[N] + Cin )
//
// Block tile: 128(M) x 64(N), 8 waves, each wave -> 16x64 (4 accumulators).
// Double-buffered LDS (one barrier per K-chunk): chunk k+1 is staged while
// chunk k feeds the WMMAs. Staging loads are unconditional (row indices are
// clamped -- garbage only lands in never-stored rows/cols; only the K-tail is
// zeroed, via mask-multiply so loads are never predicated). LDS rows padded to
// 18 floats: fragment reads are 8B-aligned ds_load_b64 on 16 distinct banks.
// ---------------------------------------------------------------------------
constexpr int LDSP = 18;                  // padded row length (even)
constexpr int ASZ  = 128 * LDSP;          // one A buffer
constexpr int BSZE = 64 * LDSP;           // one B buffer

__global__ __launch_bounds__(256)
void wmma_gemm_f32(const float* __restrict__ A, const float* __restrict__ B,
                   const float* __restrict__ Cin, float* __restrict__ Cout,
                   const float* __restrict__ bias,
                   int M, int N, int K, int lda, int ldb, int ldc, int act)
{
  __shared__ float lds_a[2 * ASZ];
  __shared__ float lds_b[2 * BSZE];

  const int tid  = threadIdx.x;
  const int lane = tid & 31;
  const int wave = tid >> 5;           // 0..7
  const int half = lane >> 4;          // 0 or 1
  const int r    = lane & 15;

  const int m0 = blockIdx.y * 128;     // block row base
  const int n0 = blockIdx.x * 64;      // block col base
  const int mw = m0 + wave * 16;       // wave row base

  // ---- accumulator init ----
  v8f acc[4];
#pragma unroll
  for (int j = 0; j < 4; ++j) {
#pragma unroll
    for (int v = 0; v < 8; ++v) {
      if (Cin) {
        int mm = mw + v + 8 * half;
        int nn = n0 + j * 16 + r;
        float val = Cin[(size_t)imin(mm, M - 1) * ldc + imin(nn, N - 1)];
        acc[j][v] = (mm < M && nn < N) ? val : 0.f;
      } else {
        acc[j][v] = 0.f;
      }
    }
  }

  // ---- precomputed staging addresses (rows clamped; no per-chunk addr math)
  const int srow = tid >> 4;   // 0..15
  const int scol = tid & 15;   // 0..15  (k offset within chunk)
  const float* aptr[8];
#pragma unroll
  for (int it = 0; it < 8; ++it)
    aptr[it] = A + (size_t)imin(m0 + srow + it * 16, M - 1) * lda;
  const float* bptr[4];
#pragma unroll
  for (int it = 0; it < 4; ++it)
    bptr[it] = B + (size_t)imin(n0 + srow + it * 16, N - 1) * ldb;
  const int sa_off = srow * LDSP + scol;   // + it*16*LDSP (immediate)

  auto stage = [&](int k0, int p) {
    float va[8], vb[4];
    if (k0 + 16 <= K) {          // block-uniform: steady state, no masks
#pragma unroll
      for (int it = 0; it < 8; ++it) va[it] = aptr[it][k0 + scol];
#pragma unroll
      for (int it = 0; it < 4; ++it) vb[it] = bptr[it][k0 + scol];
    } else {                     // K-tail: clamped index + arithmetic mask
      const int   gk = k0 + scol;
      const int   kc = gk < K ? gk : 0;
      const float mk = gk < K ? 1.f : 0.f;
#pragma unroll
      for (int it = 0; it < 8; ++it) va[it] = aptr[it][kc] * mk;
#pragma unroll
      for (int it = 0; it < 4; ++it) vb[it] = bptr[it][kc] * mk;
    }
    float* sa = &lds_a[p * ASZ + sa_off];
    float* sb = &lds_b[p * BSZE + sa_off];
#pragma unroll
    for (int it = 0; it < 8; ++it) sa[it * 16 * LDSP] = va[it];
#pragma unroll
    for (int it = 0; it < 4; ++it) sb[it * 16 * LDSP] = vb[it];
  };

  // ---- main loop: double-buffered, one barrier per chunk ----
  stage(0, 0);
  __syncthreads();
  int p = 0;
  for (int k0 = 0; k0 < K; k0 += 16) {
    if (k0 + 16 < K) stage(k0 + 16, p ^ 1);

    const float* ca = &lds_a[p * ASZ  + (wave * 16 + r) * LDSP];
    const float* cb = &lds_b[p * BSZE + r * LDSP];
#pragma unroll
    for (int ks = 0; ks < 4; ++ks) {
      const int kk = ks * 4 + 2 * half;
      const v2f a = *(const v2f*)&ca[kk];
#pragma unroll
      for (int j = 0; j < 4; ++j) {
        const v2f b = *(const v2f*)&cb[j * 16 * LDSP + kk];
        acc[j] = wmma16x16x4(a, b, acc[j]);
      }
    }
    __syncthreads();
    p ^= 1;
  }

  // ---- epilogue ----
#pragma unroll
  for (int j = 0; j < 4; ++j) {
    const int nn = n0 + j * 16 + r;
#pragma unroll
    for (int v = 0; v < 8; ++v) {
      const int mm = mw + v + 8 * half;
      if (mm < M && nn < N) {
        float val = acc[j][v];
        if (bias) val += bias[nn];
        if (act == 1) val = softplusf(val);
        Cout[(size_t)mm * ldc + nn] = val;
      }
    }
  }
}

// ---------------------------------------------------------------------------
// x (b,s,f,c,e) -> fx (b,s, (c*F+f)*E+e)
// ---------------------------------------------------------------------------
__global__ void permute_kernel(const float* __restrict__ x, float* __restrict__ fx)
{
  int i = blockIdx.x * blockDim.x + threadIdx.x;
  if (i >= ROWS * DM) return;
  int d = i % DM, row = i / DM;
  int b = row / SEQ, s = row % SEQ;
  int c = d / (FF * EE), rem = d % (FF * EE);
  int f = rem / EE, e = rem % EE;
  fx[i] = x[(((size_t)(b * SEQ + s) * FF + f) * CC + c) * EE + e];
}

// ---------------------------------------------------------------------------
// RMSNorm per row: y = x * rsqrt(mean(x^2)+eps) * w    (one block per row)
// ---------------------------------------------------------------------------
__global__ __launch_bounds__(256)
void rmsnorm_kernel(const float* __restrict__ x, const float* __restrict__ w,
                    float* __restrict__ y, int dim)
{
  __shared__ float red[256];
  const int row = blockIdx.x;
  const float* xr = x + (size_t)row * dim;
  float s = 0.f;
  for (int i = threadIdx.x; i < dim; i += 256) { float v = xr[i]; s += v * v; }
  red[threadIdx.x] = s;
  __syncthreads();
  for (int off = 128; off > 0; off >>= 1) {
    if (threadIdx.x < off) red[threadIdx.x] += red[threadIdx.x + off];
    __syncthreads();
  }
  const float rs = rsqrtf(red[0] / (float)dim + EPS);
  for (int i = threadIdx.x; i < dim; i += 256)
    y[(size_t)row * dim + i] = xr[i] * rs * w[i];
}

// ---------------------------------------------------------------------------
// Causal depthwise conv (D_CONV=4, left pad 3, S=3 unrolled) + SiLU.
// x_in is the first ED columns of xz (row stride 2*ED).
// ---------------------------------------------------------------------------
__global__ void conv_silu_kernel(const float* __restrict__ xz,
                                 const float* __restrict__ cw,
                                 const float* __restrict__ cb,
                                 float* __restrict__ xc)
{
  int t = blockIdx.x * blockDim.x + threadIdx.x;
  if (t >= BSZ * ED) return;
  int b = t / ED, e = t % ED;
  const float w1 = cw[e * 4 + 1], w2 = cw[e * 4 + 2], w3 = cw[e * 4 + 3];
  const float bias = cb[e];
  const size_t base = (size_t)(b * SEQ) * (2 * ED) + e;
  const float x0 = xz[base];
  const float x1 = xz[base + 2 * ED];
  const float x2 = xz[base + 4 * ED];
  const size_t ob = (size_t)(b * SEQ) * ED + e;
  xc[ob]           = siluf(bias + w3 * x0);
  xc[ob + ED]      = siluf(bias + w2 * x0 + w3 * x1);
  xc[ob + 2 * ED]  = siluf(bias + w1 * x0 + w2 * x1 + w3 * x2);
}

// ---------------------------------------------------------------------------
// Selective scan: per (b, e) channel, 16 states in registers, S=3 loop.
// y = (scan + D*xc) * silu(z)
// ---------------------------------------------------------------------------
__global__ void scan_kernel(const float* __restrict__ delta,
                            const float* __restrict__ dbc,
                            const float* __restrict__ xc,
                            const float* __restrict__ xz,
                            const float* __restrict__ A_log,
                            const float* __restrict__ Dp,
                            float* __restrict__ y)
{
  int t = blockIdx.x * blockDim.x + threadIdx.x;
  if (t >= BSZ * ED) return;
  int b = t / ED, e = t % ED;

  float Av[DSTATE], h[DSTATE];
#pragma unroll
  for (int n = 0; n < DSTATE; ++n) {
    Av[n] = -expf(A_log[(size_t)e * DSTATE + n]);
    h[n] = 0.f;
  }
  const float dpe = Dp[e];

  for (int s = 0; s < SEQ; ++s) {
    const size_t row = (size_t)(b * SEQ + s);
    const float d   = delta[row * ED + e];
    const float xcv = xc[row * ED + e];
    const float* dbcrow = dbc + row * DBCW;
    float acc = 0.f;
#pragma unroll
    for (int n = 0; n < DSTATE; ++n) {
      const float dA = expf(d * Av[n]);
      h[n] = dA * h[n] + d * dbcrow[DTR + n] * xcv;
      acc += h[n] * dbcrow[DTR + DSTATE + n];
    }
    acc += dpe * xcv;
    const float z = xz[row * (2 * ED) + ED + e];
    y[row * ED + e] = acc * siluf(z);
  }
}

// ---------------------------------------------------------------------------
// Tail: scalar mean/var over logits (broadcast over f collapses the BN axes)
// ---------------------------------------------------------------------------
__global__ void zero_stats_kernel(float* stats)
{
  if (threadIdx.x < 2) stats[threadIdx.x] = 0.f;
}

__global__ __launch_bounds__(256)
void reduce_stats_kernel(const float* __restrict__ x, float* stats, int n)
{
  __shared__ float s0[256], s1[256];
  float a = 0.f, b = 0.f;
  for (int i = blockIdx.x * 256 + threadIdx.x; i < n; i += gridDim.x * 256) {
    float v = x[i]; a += v; b += v * v;
  }
  s0[threadIdx.x] = a; s1[threadIdx.x] = b;
  __syncthreads();
  for (int off = 128; off > 0; off >>= 1) {
    if (threadIdx.x < off) {
      s0[threadIdx.x] += s0[threadIdx.x + off];
      s1[threadIdx.x] += s1[threadIdx.x + off];
    }
    __syncthreads();
  }
  if (threadIdx.x == 0) {
    atomicAdd(&stats[0], s0[0]);
    atomicAdd(&stats[1], s1[0]);
  }
}

__global__ void norm_logits_kernel(const float* __restrict__ logits,
                                   const float* __restrict__ stats,
                                   float* __restrict__ ln, int n)
{
  int i = blockIdx.x * blockDim.x + threadIdx.x;
  if (i >= n) return;
  const float mean = stats[0] / (float)n;
  const float var  = stats[1] / (float)n - mean * mean;
  ln[i] = (logits[i] - mean) * rsqrtf(var + EPS);
}

__global__ void rowsum_emb_kernel(const float* __restrict__ emb_w,
                                  float* __restrict__ rowsum)
{
  int o = blockIdx.x * blockDim.x + threadIdx.x;
  if (o >= OUTD) return;
  float s = 0.f;
  for (int v = 0; v < VOCAB; ++v) s += emb_w[(size_t)o * VOCAB + v];
  rowsum[o] = s;
}

// out[b,f,s,o] = bn_w[f]*ln_emb[b,s,o] + bn_b[f]*rowsum[o] + emb_b[o]
__global__ void final_out_kernel(const float* __restrict__ ln_emb,
                                 const float* __restrict__ rowsum,
                                 const float* __restrict__ bn_w,
                                 const float* __restrict__ bn_b,
                                 const float* __restrict__ emb_b,
                                 float* __restrict__ out, int total)
{
  int i = blockIdx.x * blockDim.x + threadIdx.x;
  if (i >= total) return;
  int o = i % OUTD;
  int s = (i / OUTD) % SEQ;
  int f = (i / (OUTD * SEQ)) % FF;
  int b = i / (OUTD * SEQ * FF);
  out[i] = bn_w[f] * ln_emb[(size_t)(b * SEQ + s) * OUTD + o]
         + bn_b[f] * rowsum[o] + emb_b[o];
}

// ---------------------------------------------------------------------------
// Host launcher
// ---------------------------------------------------------------------------
static inline void launch_gemm(const float* A, const float* B, const float* Cin,
                               float* Cout, const float* bias,
                               int M, int N, int K, int lda, int ldb, int ldc,
                               int act, hipStream_t stream)
{
  dim3 grid((N + 63) / 64, (M + 127) / 128);
  wmma_gemm_f32<<<grid, dim3(256), 0, stream>>>(A, B, Cin, Cout, bias,
                                                M, N, K, lda, ldb, ldc, act);
}

extern "C" void kernel_launch(void* const* d_in, const int* in_sizes, int n_in,
                              void* d_out, int out_size, void* d_ws, size_t ws_size,
                              hipStream_t stream)
{
  const float* x         = (const float*)d_in[0];
  const float* in_proj_w = (const float*)d_in[1];
  const float* conv_w    = (const float*)d_in[2];
  const float* conv_b    = (const float*)d_in[3];
  const float* x_proj_w  = (const float*)d_in[4];
  const float* dt_proj_w = (const float*)d_in[5];
  const float* dt_proj_b = (const float*)d_in[6];
  const float* A_log     = (const float*)d_in[7];
  const float* D_param   = (const float*)d_in[8];
  const float* out_proj_w= (const float*)d_in[9];
  const float* norm_w    = (const float*)d_in[10];
  const float* normf_w   = (const float*)d_in[11];
  const float* lm_head_w = (const float*)d_in[12];
  const float* bn_w      = (const float*)d_in[13];
  const float* bn_b      = (const float*)d_in[14];
  const float* emb_w     = (const float*)d_in[15];
  const float* emb_b     = (const float*)d_in[16];
  float* out = (float*)d_out;

  // Workspace carve-up (fp32 elements)
  float* ws = (float*)d_ws;
  float* h      = ws;                              // ROWS*DM
  float* xn     = h      + (size_t)ROWS * DM;      // ROWS*DM
  float* xz     = xn     + (size_t)ROWS * DM;      // ROWS*2*ED
  float* xc     = xz     + (size_t)ROWS * 2 * ED;  // ROWS*ED
  float* delta  = xc     + (size_t)ROWS * ED;      // ROWS*ED
  float* yb     = delta  + (size_t)ROWS * ED;      // ROWS*ED
  float* dbc    = yb     + (size_t)ROWS * ED;      // ROWS*DBCW
  float* logits = dbc    + (size_t)ROWS * DBCW;    // ROWS*VOCAB
  float* ln     = logits + (size_t)ROWS * VOCAB;   // ROWS*VOCAB
  float* lnemb  = ln     + (size_t)ROWS * VOCAB;   // ROWS*OUTD
  float* rowsum = lnemb  + (size_t)ROWS * OUTD;    // OUTD
  float* stats  = rowsum + OUTD;                   // 2

  const int TB = 256;

  // 0) input permute -> h
  permute_kernel<<<(ROWS * DM + TB - 1) / TB, TB, 0, stream>>>(x, h);

  // Layers
  for (int l = 0; l < 2; ++l) {
    const float* ipw = in_proj_w + (size_t)l * (2 * ED) * DM;
    const float* cw  = conv_w    + (size_t)l * ED * 4;
    const float* cb  = conv_b    + (size_t)l * ED;
    const float* xpw = x_proj_w  + (size_t)l * DBCW * ED;
    const float* dtw = dt_proj_w + (size_t)l * ED * DTR;
    const float* dtb = dt_proj_b + (size_t)l * ED;
    const float* al  = A_log     + (size_t)l * ED * DSTATE;
    const float* dp  = D_param   + (size_t)l * ED;
    const float* opw = out_proj_w+ (size_t)l * DM * ED;
    const float* nw  = norm_w    + (size_t)l * DM;

    // rmsnorm(h) -> xn
    rmsnorm_kernel<<<ROWS, TB, 0, stream>>>(h, nw, xn, DM);
    // xz = xn @ in_proj_w^T      (1536 x 6200 x 1550)
    launch_gemm(xn, ipw, nullptr, xz, nullptr,
                ROWS, 2 * ED, DM, DM, DM, 2 * ED, 0, stream);
    // causal conv + silu -> xc
    conv_silu_kernel<<<(BSZ * ED + TB - 1) / TB, TB, 0, stream>>>(xz, cw, cb, xc);
    // dbc = xc @ x_proj_w^T      (1536 x 129 x 3100)
    launch_gemm(xc, xpw, nullptr, dbc, nullptr,
                ROWS, DBCW, ED, ED, ED, DBCW, 0, stream);
    // delta = softplus(dbc[:, :97] @ dt_proj_w^T + dt_b)  (1536 x 3100 x 97)
    launch_gemm(dbc, dtw, nullptr, delta, dtb,
                ROWS, ED, DTR, DBCW, DTR, ED, 1, stream);
    // selective scan -> yb (includes D*xc and *silu(z))
    scan_kernel<<<(BSZ * ED + TB - 1) / TB, TB, 0, stream>>>(
        delta, dbc, xc, xz, al, dp, yb);
    // h += yb @ out_proj_w^T     (1536 x 1550 x 3100), fused residual
    launch_gemm(yb, opw, h, h, nullptr,
                ROWS, DM, ED, ED, ED, DM, 0, stream);
  }

  // final rmsnorm -> xn
  rmsnorm_kernel<<<ROWS, TB, 0, stream>>>(h, normf_w, xn, DM);
  // logits = xn @ lm_head_w^T   (1536 x 128 x 1550)
  launch_gemm(xn, lm_head_w, nullptr, logits, nullptr,
              ROWS, VOCAB, DM, DM, DM, VOCAB, 0, stream);

  // scalar batch-norm stats over all logits
  zero_stats_kernel<<<1, 32, 0, stream>>>(stats);
  reduce_stats_kernel<<<96, TB, 0, stream>>>(logits, stats, ROWS * VOCAB);
  norm_logits_kernel<<<(ROWS * VOCAB + TB - 1) / TB, TB, 0, stream>>>(
      logits, stats, ln, ROWS * VOCAB);

  // fold the f-broadcast: ln_emb = ln @ emb_w^T  (1536 x 256 x 128)
  rowsum_emb_kernel<<<(OUTD + TB - 1) / TB, TB, 0, stream>>>(emb_w, rowsum);
  launch_gemm(ln, emb_w, nullptr, lnemb, nullptr,
              ROWS, OUTD, VOCAB, VOCAB, VOCAB, OUTD, 0, stream);

  // out[b,f,s,o]
  final_out_kernel<<<(out_size + TB - 1) / TB, TB, 0, stream>>>(
      lnemb, rowsum, bn_w, bn_b, emb_b, out, out_size);
}